// MultiScaleAttention_45930380263463
// MI455X (gfx1250) — compile-verified
//
#include <hip/hip_runtime.h>
#include <hip/hip_bf16.h>

// ---------------------------------------------------------------------------
// MViT multi-scale attention on MI455X (gfx1250).
// All matmul stages use V_WMMA_F32_16X16X4_F32 (fp32-exact matrix pipe).
// Fragment layouts per CDNA5 ISA 7.12.2:
//   A 16x4 : lanes 0-15 -> M=lane, VGPR{0,1}=K{0,1}; lanes 16-31 -> K{2,3}
//   B 4x16 : lanes 0-15 -> N=lane, VGPR{0,1}=K{0,1}; lanes 16-31 -> K{2,3}
//   C/D    : VGPR r, lanes 0-15 -> M=r, lanes 16-31 -> M=r+8, N=lane&15
// ---------------------------------------------------------------------------

typedef float v2f __attribute__((ext_vector_type(2)));
typedef float v8f __attribute__((ext_vector_type(8)));

#define B_      8
#define NH_     8
#define HD_     64
#define DIM_    512
#define N_TOK   3137           // 56*56 + 1
#define NQ      3137
#define NK      785            // 28*28 + 1
#define MROWS   (B_ * N_TOK)   // 25096
#define SCALE_  0.125f         // 64^-0.5
#define EPS_    1e-6f
#define SKP     800            // padded key dim (multiple of 16)

__device__ __forceinline__ v8f wmma_f32(v2f a, v2f b, v8f c) {
  return __builtin_amdgcn_wmma_f32_16x16x4_f32(
      /*neg_a=*/false, a, /*neg_b=*/false, b,
      /*c_mod=*/(short)0, c, /*reuse_a=*/false, /*reuse_b=*/false);
}

// ---------------------------------------------------------------------------
// 1) QKV GEMM: C[m, c] = x[m,:] . qkv_w[c,:] + qkv_b[c], scattered to
//    q/k/v raw buffers laid out [B, NH, N_TOK, HD].
// ---------------------------------------------------------------------------
__global__ void qkv_gemm_kernel(const float* __restrict__ x,
                                const float* __restrict__ w,
                                const float* __restrict__ bias,
                                float* __restrict__ q_raw,
                                float* __restrict__ k_raw,
                                float* __restrict__ v_raw) {
  const int lane = threadIdx.x & 31;
  const int wave = threadIdx.x >> 5;
  const int MT = (MROWS + 15) / 16;   // 1569
  const int NT = 1536 / 16;           // 96
  long tile = (long)blockIdx.x * 8 + wave;
  if (tile >= (long)MT * NT) return;
  const int tm = (int)(tile % MT);
  const int tn = (int)(tile / MT);
  const int m0 = tm * 16, n0 = tn * 16;

  const int half2 = (lane >> 4) * 2;       // K sub-offset for this half-wave
  int am = m0 + (lane & 15);
  if (am >= MROWS) am = MROWS - 1;
  const float* arow = x + (long)am * DIM_;
  const float* brow = w + (long)(n0 + (lane & 15)) * DIM_;  // B^T column

  v8f acc = {};
  for (int k0 = 0; k0 < DIM_; k0 += 4) {
    v2f a, b;
    a[0] = arow[k0 + half2];
    a[1] = arow[k0 + half2 + 1];
    b[0] = brow[k0 + half2];
    b[1] = brow[k0 + half2 + 1];
    acc = wmma_f32(a, b, acc);
  }

  const int col = lane & 15;
  const int c = n0 + col;
  const int which = c / DIM_;       // 0=q 1=k 2=v
  const int rem = c % DIM_;
  const int h = rem / HD_;
  const int d = rem % HD_;
  const float bv = bias[c];
  float* dst = (which == 0) ? q_raw : (which == 1) ? k_raw : v_raw;
#pragma unroll
  for (int r = 0; r < 8; ++r) {
    int row = m0 + r + ((lane >> 4) * 8);
    if (row < MROWS) {
      int bidx = row / N_TOK;
      int nidx = row % N_TOK;
      dst[(((long)bidx * NH_ + h) * N_TOK + nidx) * HD_ + d] = acc[r] + bv;
    }
  }
}

// ---------------------------------------------------------------------------
// 2) Depthwise 3x3 conv pool (pad=1, stride s) + LayerNorm over HD.
//    src [64, N_TOK, HD] per (b*NH), dst [64, oh*ow+1, HD]. cls row passes
//    through conv untouched; everything gets LN.
// ---------------------------------------------------------------------------
__global__ void pool_kernel(const float* __restrict__ src,
                            float* __restrict__ dst,
                            const float* __restrict__ cw,   // [HD,1,3,3]
                            const float* __restrict__ lw,
                            const float* __restrict__ lb,
                            int stride, int oh, int ow) {
  const int bh = blockIdx.y;
  const int nout = oh * ow + 1;
  const int p = blockIdx.x * blockDim.x + threadIdx.x;
  if (p >= nout) return;
  const float* base = src + (long)bh * N_TOK * HD_;

  float acc[HD_];
  if (p == 0) {
#pragma unroll
    for (int c = 0; c < HD_; ++c) acc[c] = base[c];
  } else {
#pragma unroll
    for (int c = 0; c < HD_; ++c) acc[c] = 0.f;
    const int oy = (p - 1) / ow, ox = (p - 1) % ow;
    for (int dy = 0; dy < 3; ++dy) {
      int iy = oy * stride + dy - 1;
      if (iy < 0 || iy >= 56) continue;
      for (int dx = 0; dx < 3; ++dx) {
        int ix = ox * stride + dx - 1;
        if (ix < 0 || ix >= 56) continue;
        const float* in = base + (long)(1 + iy * 56 + ix) * HD_;
        const int t = dy * 3 + dx;
#pragma unroll
        for (int c = 0; c < HD_; ++c) acc[c] += in[c] * cw[c * 9 + t];
      }
    }
  }
  float m = 0.f;
#pragma unroll
  for (int c = 0; c < HD_; ++c) m += acc[c];
  m *= (1.f / HD_);
  float var = 0.f;
#pragma unroll
  for (int c = 0; c < HD_; ++c) { float t = acc[c] - m; var += t * t; }
  var *= (1.f / HD_);
  const float inv = rsqrtf(var + EPS_);
  float* out = dst + ((long)bh * nout + p) * HD_;
#pragma unroll
  for (int c = 0; c < HD_; ++c) out[c] = (acc[c] - m) * inv * lw[c] + lb[c];
}

// ---------------------------------------------------------------------------
// 3) Attention: per (b,h) and 16-query tile. S = scale*q@k^T + decomposed
//    rel-pos bias, softmax over 785 keys, O = P@v + residual(q).
// ---------------------------------------------------------------------------
__global__ __launch_bounds__(32) void attn_kernel(
    const float* __restrict__ qp,    // [64, NQ, HD]
    const float* __restrict__ kp,    // [64, NK, HD]
    const float* __restrict__ vp,    // [64, NK, HD]
    const float* __restrict__ rph,   // [111, HD]
    const float* __restrict__ rpw,   // [111, HD]
    float* __restrict__ out) {       // [64, NQ, HD]
  __shared__ float qs[16][HD_];
  __shared__ float rh[16][28];
  __shared__ float rw[16][28];
  __shared__ float S[16][SKP];

  const int lane = threadIdx.x;
  const int t = blockIdx.x;       // query tile
  const int bh = blockIdx.y;
  const float* qb = qp + (long)bh * NQ * HD_;
  const float* kb = kp + (long)bh * NK * HD_;
  const float* vb = vp + (long)bh * NK * HD_;

  // stage 16x64 q tile in LDS
  for (int idx = lane; idx < 16 * HD_; idx += 32) {
    int r = idx / HD_, c = idx % HD_;
    int n = t * 16 + r;
    if (n >= NQ) n = NQ - 1;
    qs[r][c] = qb[(long)n * HD_ + c];
  }
  __syncthreads();

  // rel-pos partial biases: rh[r][kh] = q_r . rel_pos_h[y - 2*kh + 54]
  for (int idx = lane; idx < 16 * 28; idx += 32) {
    int r = idx / 28, kj = idx % 28;
    int n = t * 16 + r;
    float sh = 0.f, sw = 0.f;
    if (n >= 1 && n < NQ) {
      int y = (n - 1) / 56, x = (n - 1) % 56;
      const float* Rh = rph + (long)(y - 2 * kj + 54) * HD_;
      const float* Rw = rpw + (long)(x - 2 * kj + 54) * HD_;
#pragma unroll 8
      for (int c = 0; c < HD_; ++c) {
        sh += qs[r][c] * Rh[c];
        sw += qs[r][c] * Rw[c];
      }
    }
    rh[r][kj] = sh;
    rw[r][kj] = sw;
  }
  __syncthreads();

  const int ar = lane & 15;
  const int half2 = (lane >> 4) * 2;

  // S = scale * q @ k^T + bias   (50 key tiles of 16)
  for (int kt = 0; kt < SKP / 16; ++kt) {
    v8f acc = {};
    int jb = kt * 16 + (lane & 15);
    int jc = jb < NK ? jb : NK - 1;
    const float* krow = kb + (long)jc * HD_;
    for (int k4 = 0; k4 < HD_ / 4; ++k4) {
      int kk = k4 * 4 + half2;
      v2f a, b;
      a[0] = qs[ar][kk];  a[1] = qs[ar][kk + 1];
      b[0] = krow[kk];    b[1] = krow[kk + 1];
      acc = wmma_f32(a, b, acc);
    }
    const int col = lane & 15;
    const int j = kt * 16 + col;
#pragma unroll
    for (int r = 0; r < 8; ++r) {
      int row = r + ((lane >> 4) * 8);
      int qn = t * 16 + row;
      float s;
      if (j < NK) {
        s = acc[r] * SCALE_;
        if (j >= 1 && qn >= 1 && qn < NQ) {
          int kh = (j - 1) / 28, kw = (j - 1) % 28;
          s += rh[row][kh] + rw[row][kw];
        }
      } else {
        s = -1e30f;   // padded keys
      }
      S[row][j] = s;
    }
  }
  __syncthreads();

  // row softmax over 800 (padded entries vanish)
  if (lane < 16) {
    float mx = -1e30f;
    for (int j = 0; j < SKP; ++j) mx = fmaxf(mx, S[lane][j]);
    float sum = 0.f;
    for (int j = 0; j < SKP; ++j) {
      float e = __expf(S[lane][j] - mx);
      S[lane][j] = e;
      sum += e;
    }
    float inv = 1.f / sum;
    for (int j = 0; j < SKP; ++j) S[lane][j] *= inv;
  }
  __syncthreads();

  // O = P @ v (+ residual q for non-cls rows)
  const int bn = lane & 15;
  for (int nt = 0; nt < HD_ / 16; ++nt) {
    v8f acc = {};
    const int d = nt * 16 + bn;
    for (int k4 = 0; k4 < SKP / 4; ++k4) {
      int kk = k4 * 4 + half2;
      v2f a, b;
      a[0] = S[ar][kk];  a[1] = S[ar][kk + 1];
      int kr0 = kk;      if (kr0 >= NK) kr0 = NK - 1;   // P==0 there
      int kr1 = kk + 1;  if (kr1 >= NK) kr1 = NK - 1;
      b[0] = vb[(long)kr0 * HD_ + d];
      b[1] = vb[(long)kr1 * HD_ + d];
      acc = wmma_f32(a, b, acc);
    }
    const int col = lane & 15;
#pragma unroll
    for (int r = 0; r < 8; ++r) {
      int row = r + ((lane >> 4) * 8);
      int qn = t * 16 + row;
      if (qn < NQ) {
        int dd = nt * 16 + col;
        float val = acc[r];
        if (qn >= 1) val += qs[row][dd];
        out[((long)bh * NQ + qn) * HD_ + dd] = val;
      }
    }
  }
}

// ---------------------------------------------------------------------------
// 4) Output projection: gathers A from [B, NH, NQ, HD] (head transpose) and
//    writes final [B, N, 512].
// ---------------------------------------------------------------------------
__global__ void proj_gemm_kernel(const float* __restrict__ a,
                                 const float* __restrict__ w,
                                 const float* __restrict__ bias,
                                 float* __restrict__ out) {
  const int lane = threadIdx.x & 31;
  const int wave = threadIdx.x >> 5;
  const int MT = (MROWS + 15) / 16;
  const int NT = DIM_ / 16;   // 32
  long tile = (long)blockIdx.x * 8 + wave;
  if (tile >= (long)MT * NT) return;
  const int tm = (int)(tile % MT);
  const int tn = (int)(tile / MT);
  const int m0 = tm * 16, n0 = tn * 16;

  const int half2 = (lane >> 4) * 2;
  int am = m0 + (lane & 15);
  if (am >= MROWS) am = MROWS - 1;
  const int bidx = am / N_TOK, nidx = am % N_TOK;
  // element (am, c) lives at abase + (c/64)*NQ*HD + (c%64)
  const float* abase = a + ((long)bidx * NH_) * (long)N_TOK * HD_ + (long)nidx * HD_;
  const float* brow = w + (long)(n0 + (lane & 15)) * DIM_;

  v8f acc = {};
  for (int k0 = 0; k0 < DIM_; k0 += 4) {
    int c0 = k0 + half2;
    int c1 = c0 + 1;
    v2f a2, b2;
    a2[0] = abase[(long)(c0 >> 6) * (N_TOK * HD_) + (c0 & 63)];
    a2[1] = abase[(long)(c1 >> 6) * (N_TOK * HD_) + (c1 & 63)];
    b2[0] = brow[c0];
    b2[1] = brow[c1];
    acc = wmma_f32(a2, b2, acc);
  }
  const int col = lane & 15;
  const float bv = bias[n0 + col];
#pragma unroll
  for (int r = 0; r < 8; ++r) {
    int row = m0 + r + ((lane >> 4) * 8);
    if (row < MROWS) out[(long)row * DIM_ + n0 + col] = acc[r] + bv;
  }
}

// ---------------------------------------------------------------------------
extern "C" void kernel_launch(void* const* d_in, const int* in_sizes, int n_in,
                              void* d_out, int out_size, void* d_ws, size_t ws_size,
                              hipStream_t stream) {
  const float* x      = (const float*)d_in[0];
  const float* qkv_w  = (const float*)d_in[1];
  const float* qkv_b  = (const float*)d_in[2];
  const float* proj_w = (const float*)d_in[3];
  const float* proj_b = (const float*)d_in[4];
  const float* pqw    = (const float*)d_in[5];
  const float* pkw    = (const float*)d_in[6];
  const float* pvw    = (const float*)d_in[7];
  const float* lnq_w  = (const float*)d_in[8];
  const float* lnq_b  = (const float*)d_in[9];
  const float* lnk_w  = (const float*)d_in[10];
  const float* lnk_b  = (const float*)d_in[11];
  const float* lnv_w  = (const float*)d_in[12];
  const float* lnv_b  = (const float*)d_in[13];
  const float* rph    = (const float*)d_in[14];
  const float* rpw    = (const float*)d_in[15];
  (void)in_sizes; (void)n_in; (void)out_size; (void)ws_size;

  float* ws = (float*)d_ws;
  const long SZ_RAW = (long)B_ * NH_ * N_TOK * HD_;   // 12,849,152 floats
  const long SZ_KP  = (long)B_ * NH_ * NK * HD_;      //  3,215,360 floats
  float* q_raw  = ws;
  float* k_raw  = q_raw + SZ_RAW;
  float* v_raw  = k_raw + SZ_RAW;
  float* q_pool = v_raw + SZ_RAW;
  float* k_pool = q_pool + SZ_RAW;
  float* v_pool = k_pool + SZ_KP;
  float* attn_o = q_raw;   // q_raw dead after pool_q; reuse for attention out

  { // 1) QKV GEMM
    long tiles = (long)((MROWS + 15) / 16) * (1536 / 16);
    int blocks = (int)((tiles + 7) / 8);
    qkv_gemm_kernel<<<blocks, 256, 0, stream>>>(x, qkv_w, qkv_b,
                                                q_raw, k_raw, v_raw);
  }
  { // 2) pools + LN
    dim3 gq((N_TOK + 255) / 256, B_ * NH_);
    pool_kernel<<<gq, 256, 0, stream>>>(q_raw, q_pool, pqw, lnq_w, lnq_b, 1, 56, 56);
    dim3 gk((NK + 255) / 256, B_ * NH_);
    pool_kernel<<<gk, 256, 0, stream>>>(k_raw, k_pool, pkw, lnk_w, lnk_b, 2, 28, 28);
    pool_kernel<<<gk, 256, 0, stream>>>(v_raw, v_pool, pvw, lnv_w, lnv_b, 2, 28, 28);
  }
  { // 3) attention (score + bias + softmax + PV + residual)
    dim3 g((NQ + 15) / 16, B_ * NH_);
    attn_kernel<<<g, 32, 0, stream>>>(q_pool, k_pool, v_pool, rph, rpw, attn_o);
  }
  { // 4) output projection
    long tiles = (long)((MROWS + 15) / 16) * (DIM_ / 16);
    int blocks = (int)((tiles + 7) / 8);
    proj_gemm_kernel<<<blocks, 256, 0, stream>>>(attn_o, proj_w, proj_b,
                                                 (float*)d_out);
  }
}